// RGCN_ComplEx_84275848282313
// MI455X (gfx1250) — compile-verified
//
#include <hip/hip_runtime.h>
#include <hip/hip_bf16.h>

// ---------------------------------------------------------------------------
// RGCN (2 layers, basis decomposition, NB=2) + ComplEx scoring for gfx1250.
//
//  * scatter phase: memory/atomic bound -> wave-per-edge, float4 gathers from
//    L2-resident x, global_atomic_add_f32 into per-basis accumulators.
//  * dense phase: one fused f32 WMMA GEMM per layer:
//        accA = [agg0 | agg1] (N x 256) @ [B0; B1]
//        accX = x (N x 128) @ root
//        out  = accA * invdeg[row] + accX + bias   (row scaling commutes with
//                                                   the right matmul)
//    V_WMMA_F32_16X16X4_F32; A-tile staged in LDS via async-to-LDS B128 DMA
//    (ASYNCcnt) when the builtin is available; 8 waves/block, each owning a
//    16-wide column tile, 96 WMMA per tile (K=384).
//  * ComplEx: wave-per-triple, shuffle reduction.
// ---------------------------------------------------------------------------

typedef float v2f __attribute__((ext_vector_type(2)));
typedef float v8f __attribute__((ext_vector_type(8)));
typedef int i32x4 __attribute__((ext_vector_type(4)));

#define DD 128  // node feature dim (2*H)

#if defined(__has_builtin)
#if __has_builtin(__builtin_amdgcn_global_load_async_to_lds_b128) && \
    __has_builtin(__builtin_amdgcn_s_wait_asynccnt)
#define USE_ASYNC_STAGE 1
#endif
#endif
#ifndef USE_ASYNC_STAGE
#define USE_ASYNC_STAGE 0
#endif

// Builtin signature (from clang diagnostic): param0 is i32x4 in the global
// (__device__/AS1) address space; param1 the LDS (AS3) side.
#define AS_GLB_V4(p) ((__attribute__((address_space(1))) i32x4*)(p))
#define AS_LDS_V4(p) ((__attribute__((address_space(3))) i32x4*)(p))

// ---------------------------------------------------------------- zero fill
__global__ void __launch_bounds__(256) zero_f32_kernel(float* __restrict__ p, int n) {
  int i = blockIdx.x * blockDim.x + threadIdx.x;
  int stride = gridDim.x * blockDim.x;
  for (; i < n; i += stride) p[i] = 0.0f;
}

// ---------------------------------------------------------------- degree
__global__ void __launch_bounds__(256) degree_kernel(const int* __restrict__ dst,
                                                     float* __restrict__ deg, int E) {
  int e = blockIdx.x * blockDim.x + threadIdx.x;
  if (e < E) atomicAdd(&deg[dst[e]], 1.0f);
}

__global__ void __launch_bounds__(256) invdeg_kernel(const float* __restrict__ deg,
                                                     float* __restrict__ invdeg, int n) {
  int i = blockIdx.x * blockDim.x + threadIdx.x;
  if (i < n) invdeg[i] = 1.0f / fmaxf(deg[i], 1.0f);
}

// ---------------------------------------------------------------- scatter
// One wave (32 lanes) per edge; lane l owns dims [4l, 4l+4).
__global__ void __launch_bounds__(256) scatter_kernel(const float* __restrict__ x,
                                                      const int* __restrict__ src,
                                                      const int* __restrict__ dst,
                                                      const int* __restrict__ et,
                                                      const float* __restrict__ comp, // (R,2)
                                                      float* __restrict__ agg0,
                                                      float* __restrict__ agg1, int E) {
  int gid  = blockIdx.x * blockDim.x + threadIdx.x;
  int edge = gid >> 5;
  int lane = gid & 31;
  if (edge >= E) return;
  int s = src[edge];
  int d = dst[edge];
  int r = et[edge];
  float c0 = comp[r * 2 + 0];
  float c1 = comp[r * 2 + 1];
  const float4* xp = (const float4*)(x + (size_t)s * DD);
  float4 v = xp[lane];
  float* p0 = agg0 + (size_t)d * DD + lane * 4;
  float* p1 = agg1 + (size_t)d * DD + lane * 4;
  atomicAdd(p0 + 0, v.x * c0);
  atomicAdd(p0 + 1, v.y * c0);
  atomicAdd(p0 + 2, v.z * c0);
  atomicAdd(p0 + 3, v.w * c0);
  atomicAdd(p1 + 0, v.x * c1);
  atomicAdd(p1 + 1, v.y * c1);
  atomicAdd(p1 + 2, v.z * c1);
  atomicAdd(p1 + 3, v.w * c1);
}

// ---------------------------------------------------------------- fused GEMM
// out[16 rows x 128 cols] per block; wave w -> columns [16w, 16w+16).
// A (16 x 384) = [agg0 | agg1 | x], staged in LDS (pure copy -> async DMA).
// W (384 x 128) = [bases(2x128x128 contiguous); root(128x128)].
__global__ void __launch_bounds__(256) rgcn_gemm_kernel(const float* __restrict__ agg0,
                                                        const float* __restrict__ agg1,
                                                        const float* __restrict__ xin,
                                                        const float* __restrict__ invdeg,
                                                        const float* __restrict__ bases, // 256x128
                                                        const float* __restrict__ root,  // 128x128
                                                        const float* __restrict__ bias,  // 128
                                                        float* __restrict__ out,
                                                        int n, int relu) {
  __shared__ float Alds[16][388];  // +4 pad: row stride hits distinct banks

  const int tid     = threadIdx.x;
  const int rowBase = blockIdx.x * 16;

  // ---- Stage A-tile: pure B128 copies of [agg0 | agg1 | x] rows ----------
  {
    const float* srcs[3] = {agg0, agg1, xin};
#pragma unroll
    for (int s = 0; s < 3; ++s) {
      const float* sp = srcs[s];
#pragma unroll
      for (int i = 0; i < 2; ++i) {
        int linear = tid + 256 * i;   // 512 float4 chunks per 16x128 block
        int row = linear >> 5;        // 32 float4 per row
        int c4  = (linear & 31) * 4;  // float column within the row
        int g   = rowBase + row;
        float* lp = &Alds[row][s * 128 + c4];
        if (g < n) {
          const float* gp = sp + (size_t)g * DD + c4;
#if USE_ASYNC_STAGE
          __builtin_amdgcn_global_load_async_to_lds_b128(
              AS_GLB_V4(const_cast<float*>(gp)), AS_LDS_V4(lp), 0, 0);
#else
          *(float4*)lp = *(const float4*)gp;
#endif
        } else {
          float4 z = {0.0f, 0.0f, 0.0f, 0.0f};
          *(float4*)lp = z;
        }
      }
    }
#if USE_ASYNC_STAGE
    __builtin_amdgcn_s_wait_asynccnt(0);
#endif
  }
  __syncthreads();

  const int wave    = tid >> 5;   // 0..7 column tile
  const int lane    = tid & 31;
  const int half    = lane >> 4;  // K sub-pair select per ISA A/B layout
  const int l16     = lane & 15;
  const int colBase = wave * 16;

  v8f accA = {};  // basis contribution (scaled by invdeg in epilogue)
  v8f accX = {};  // self-loop (root) contribution

  // K = 0..255 : basis matrices -> accA
#pragma unroll 4
  for (int k0 = 0; k0 < 256; k0 += 4) {
    int ka = k0 + 2 * half;
    v2f a, b;
    a.x = Alds[l16][ka];
    a.y = Alds[l16][ka + 1];
    b.x = bases[ka * DD + colBase + l16];
    b.y = bases[(ka + 1) * DD + colBase + l16];
    accA = __builtin_amdgcn_wmma_f32_16x16x4_f32(false, a, false, b, (short)0, accA,
                                                 false, false);
  }
  // K = 256..383 : root (self-loop) matrix -> accX
#pragma unroll 4
  for (int k0 = 0; k0 < 128; k0 += 4) {
    int ka = k0 + 2 * half;
    v2f a, b;
    a.x = Alds[l16][256 + ka];
    a.y = Alds[l16][256 + ka + 1];
    b.x = root[ka * DD + colBase + l16];
    b.y = root[(ka + 1) * DD + colBase + l16];
    accX = __builtin_amdgcn_wmma_f32_16x16x4_f32(false, a, false, b, (short)0, accX,
                                                 false, false);
  }

  float bv = bias[colBase + l16];
#pragma unroll
  for (int r = 0; r < 8; ++r) {
    int row = rowBase + r + 8 * half;  // C/D layout: VGPR r -> M = r + 8*half
    if (row < n) {
      float v = accA[r] * invdeg[row] + accX[r] + bv;
      if (relu) v = fmaxf(v, 0.0f);
      out[(size_t)row * DD + colBase + l16] = v;
    }
  }
}

// ---------------------------------------------------------------- ComplEx
// One wave per triple; lane owns 2 of the 64 complex dims.
__global__ void __launch_bounds__(256) complex_score_kernel(const float* __restrict__ x,
                                                            const float* __restrict__ rel,
                                                            const int* __restrict__ h_idx,
                                                            const int* __restrict__ r_idx,
                                                            const int* __restrict__ t_idx,
                                                            float* __restrict__ out, int T) {
  int gid  = blockIdx.x * blockDim.x + threadIdx.x;
  int trip = gid >> 5;
  int lane = gid & 31;
  if (trip >= T) return;
  int h = h_idx[trip];
  int r = r_idx[trip];
  int t = t_idx[trip];
  const float2* hp = (const float2*)(x + (size_t)h * DD);
  const float2* tp = (const float2*)(x + (size_t)t * DD);
  const float2* rp = (const float2*)(rel + (size_t)r * DD);
  float2 hre = hp[lane], him = hp[lane + 32];
  float2 tre = tp[lane], tim = tp[lane + 32];
  float2 rre = rp[lane], rim = rp[lane + 32];
  float s = hre.x * rre.x * tre.x + hre.x * rim.x * tim.x + him.x * rre.x * tim.x -
            him.x * rim.x * tre.x + hre.y * rre.y * tre.y + hre.y * rim.y * tim.y +
            him.y * rre.y * tim.y - him.y * rim.y * tre.y;
  for (int off = 16; off > 0; off >>= 1) s += __shfl_down(s, off, 32);
  if (lane == 0) out[trip] = s;
}

// ---------------------------------------------------------------- launcher
extern "C" void kernel_launch(void* const* d_in, const int* in_sizes, int n_in,
                              void* d_out, int out_size, void* d_ws, size_t ws_size,
                              hipStream_t stream) {
  const int* edge_index = (const int*)d_in[0];
  const int* edge_type  = (const int*)d_in[1];
  const int* h_idx      = (const int*)d_in[2];
  const int* r_idx      = (const int*)d_in[3];
  const int* t_idx      = (const int*)d_in[4];
  const float* node_emb = (const float*)d_in[5];
  const float* rel_emb  = (const float*)d_in[6];
  const float* comp1    = (const float*)d_in[7];
  const float* bases1   = (const float*)d_in[8];
  const float* root1    = (const float*)d_in[9];
  const float* bias1    = (const float*)d_in[10];
  const float* comp2    = (const float*)d_in[11];
  const float* bases2   = (const float*)d_in[12];
  const float* root2    = (const float*)d_in[13];
  const float* bias2    = (const float*)d_in[14];

  const int E = in_sizes[1];
  const int T = in_sizes[2];
  const int N = in_sizes[5] / DD;

  const int* src = edge_index;      // edge_index[0]
  const int* dst = edge_index + E;  // edge_index[1]

  // Workspace layout (floats): deg[N] invdeg[N] agg0[N*128] agg1[N*128]
  //                            x1[N*128] x2[N*128]  (~206 MB total)
  float* ws     = (float*)d_ws;
  float* deg    = ws;
  float* invdeg = deg + N;
  float* agg0   = invdeg + N;
  float* agg1   = agg0 + (size_t)N * DD;
  float* x1     = agg1 + (size_t)N * DD;
  float* x2     = x1 + (size_t)N * DD;

  const int ZB = 2048;  // blocks for grid-stride zero fill
  const int nAgg = N * DD;

  // degrees (same for both layers)
  zero_f32_kernel<<<ZB, 256, 0, stream>>>(deg, N);
  degree_kernel<<<(E + 255) / 256, 256, 0, stream>>>(dst, deg, E);
  invdeg_kernel<<<(N + 255) / 256, 256, 0, stream>>>(deg, invdeg, N);

  const int scatterBlocks = (int)(((long long)E * 32 + 255) / 256);
  const int gemmBlocks    = (N + 15) / 16;

  // ---- layer 1 ----
  zero_f32_kernel<<<ZB, 256, 0, stream>>>(agg0, nAgg);
  zero_f32_kernel<<<ZB, 256, 0, stream>>>(agg1, nAgg);
  scatter_kernel<<<scatterBlocks, 256, 0, stream>>>(node_emb, src, dst, edge_type, comp1,
                                                    agg0, agg1, E);
  rgcn_gemm_kernel<<<gemmBlocks, 256, 0, stream>>>(agg0, agg1, node_emb, invdeg, bases1,
                                                   root1, bias1, x1, N, /*relu=*/1);

  // ---- layer 2 ----
  zero_f32_kernel<<<ZB, 256, 0, stream>>>(agg0, nAgg);
  zero_f32_kernel<<<ZB, 256, 0, stream>>>(agg1, nAgg);
  scatter_kernel<<<scatterBlocks, 256, 0, stream>>>(x1, src, dst, edge_type, comp2, agg0,
                                                    agg1, E);
  rgcn_gemm_kernel<<<gemmBlocks, 256, 0, stream>>>(agg0, agg1, x1, invdeg, bases2, root2,
                                                   bias2, x2, N, /*relu=*/0);

  // ---- ComplEx scoring ----
  const int scoreBlocks = (int)(((long long)T * 32 + 255) / 256);
  complex_score_kernel<<<scoreBlocks, 256, 0, stream>>>(x2, rel_emb, h_idx, r_idx, t_idx,
                                                        (float*)d_out, T);
}